// GSA_69063074119914
// MI455X (gfx1250) — compile-verified
//
#include <hip/hip_runtime.h>

typedef __attribute__((ext_vector_type(2))) float v2f;
typedef __attribute__((ext_vector_type(8))) float v8f;

#define BATCH 16
#define HW    16384   // 128*128 queries per batch
#define NKV   256     // 16*16 downsampled keys/values per batch

// ---------------------------------------------------------------------------
// Kernel 1: per batch, compute down[j] via 8x8 stride-8 conv, then
// k_j = down*wk+bk, v_j = down*wv+bv (stored interleaved), and block-reduce
// kmin/kmax. One block per batch, 256 threads (thread j owns patch j).
// ---------------------------------------------------------------------------
__global__ __launch_bounds__(256)
void gsa_prep_kernel(const float* __restrict__ x,
                     const float* __restrict__ wk_p, const float* __restrict__ bk_p,
                     const float* __restrict__ wv_p, const float* __restrict__ bv_p,
                     const float* __restrict__ conv_w, const float* __restrict__ conv_b,
                     float* __restrict__ ws_kv,       // [B][256][2] interleaved (k,v)
                     float* __restrict__ ws_kmin, float* __restrict__ ws_kmax) {
    const int b = blockIdx.x;
    const int t = threadIdx.x;           // j in [0,256)

    __shared__ float scw[64];
    __shared__ float smin[256];
    __shared__ float smax[256];

    if (t < 64) scw[t] = conv_w[t];
    __syncthreads();

    const int ph = t >> 4;               // patch row (0..15)
    const int pw = t & 15;               // patch col (0..15)
    const float* xp = x + (size_t)b * HW + (ph * 8) * 128 + pw * 8;

    float sum = 0.0f;
    #pragma unroll
    for (int r = 0; r < 8; ++r) {
        #pragma unroll
        for (int c = 0; c < 8; ++c) {
            sum = fmaf(xp[r * 128 + c], scw[r * 8 + c], sum);
        }
    }
    const float down = sum + conv_b[0];
    const float kv_k = fmaf(down, wk_p[0], bk_p[0]);
    const float kv_v = fmaf(down, wv_p[0], bv_p[0]);

    ws_kv[(size_t)b * (2 * NKV) + 2 * t]     = kv_k;
    ws_kv[(size_t)b * (2 * NKV) + 2 * t + 1] = kv_v;

    smin[t] = kv_k;
    smax[t] = kv_k;
    __syncthreads();
    #pragma unroll
    for (int s = 128; s > 0; s >>= 1) {
        if (t < s) {
            smin[t] = fminf(smin[t], smin[t + s]);
            smax[t] = fmaxf(smax[t], smax[t + s]);
        }
        __syncthreads();
    }
    if (t == 0) {
        ws_kmin[b] = smin[0];
        ws_kmax[b] = smax[0];
    }
}

// ---------------------------------------------------------------------------
// Kernel 2: softmax attention via V_WMMA_F32_16X16X4_F32.
// Each wave (32 lanes) owns 16 queries. A = 16x4 exp-weight tile,
// B col0 = v chunk, col1 = ones -> D col0 = numerator, col1 = denominator.
// 64 chunks of K=4 cover all 256 keys; f32 accumulation throughout.
// Inner loop: one ds_load_b128 (interleaved k/v), arithmetic lane-select for B
// (no EXEC divergence -> EXEC all-1s as WMMA requires), one WMMA per chunk.
// Block = 256 threads = 8 waves = 128 queries. Grid = (HW/128, BATCH).
// ---------------------------------------------------------------------------
__global__ __launch_bounds__(256)
void gsa_attn_kernel(const float* __restrict__ x,
                     const float* __restrict__ wq_p, const float* __restrict__ bq_p,
                     const float* __restrict__ ws_kv,
                     const float* __restrict__ ws_kmin, const float* __restrict__ ws_kmax,
                     float* __restrict__ out) {
    const int b = blockIdx.y;
    const int t = threadIdx.x;
    const int wave = t >> 5;
    const int lane = t & 31;

    __shared__ __align__(16) float skv[2 * NKV];   // interleaved (k_j, v_j)
    {
        const float2 p = ((const float2*)(ws_kv + (size_t)b * (2 * NKV)))[t];
        ((float2*)skv)[t] = p;
    }
    __syncthreads();

    const int qbase = blockIdx.x * 128 + wave * 16;   // first of 16 queries for this wave
    const int m = lane & 15;                          // row M this lane feeds in A
    const int h = lane >> 4;                          // K-half select (0: K=0,1  1: K=2,3)

    const float q  = fmaf(x[(size_t)b * HW + qbase + m], wq_p[0], bq_p[0]);
    const float mx = fmaxf(q * ws_kmax[b], q * ws_kmin[b]);  // max_j q*k_j, branch-free

    // Per-lane constants for building B: col0 = v, col1 = 1, others 0.
    const float csel = (m == 0) ? 1.0f : 0.0f;
    const float cone = (m == 1) ? 1.0f : 0.0f;

    v8f acc = {};
    #pragma unroll 4
    for (int c = 0; c < NKV / 4; ++c) {
        const int j = 4 * c + 2 * h;                  // first of this lane's two K slots
        // (k[j], v[j], k[j+1], v[j+1]) -- 16B aligned since j is even
        const float4 kv = *(const float4*)&skv[2 * j];

        v2f a;
        a.x = __expf(fmaf(q, kv.x, -mx));
        a.y = __expf(fmaf(q, kv.z, -mx));

        v2f bb;
        bb.x = fmaf(kv.y, csel, cone);
        bb.y = fmaf(kv.w, csel, cone);

        // D = A(16x4 f32) * B(4x16 f32) + C ; full f32 accumulation
        acc = __builtin_amdgcn_wmma_f32_16x16x4_f32(
            /*neg_a=*/false, a, /*neg_b=*/false, bb,
            /*c_mod=*/(short)0, acc, /*reuse_a=*/false, /*reuse_b=*/false);
    }

    // D layout: VGPR r, lanes 0-15 -> M=r (N=lane), lanes 16-31 -> M=8+r.
    // Numerator lives at N=0 (lanes 0,16); denominator at N=1 (lanes 1,17).
    #pragma unroll
    for (int r = 0; r < 8; ++r) {
        const float num = acc[r];
        const float den = __shfl_down(acc[r], 1, 32);   // lane0<-lane1, lane16<-lane17
        if (m == 0) {
            out[(size_t)b * HW + qbase + 8 * h + r] = num / den;
        }
    }
}

extern "C" void kernel_launch(void* const* d_in, const int* in_sizes, int n_in,
                              void* d_out, int out_size, void* d_ws, size_t ws_size,
                              hipStream_t stream) {
    (void)in_sizes; (void)n_in; (void)out_size; (void)ws_size;

    const float* x      = (const float*)d_in[0];
    const float* wq     = (const float*)d_in[1];
    const float* bq     = (const float*)d_in[2];
    const float* wk     = (const float*)d_in[3];
    const float* bk     = (const float*)d_in[4];
    const float* wv     = (const float*)d_in[5];
    const float* bv     = (const float*)d_in[6];
    const float* conv_w = (const float*)d_in[7];
    const float* conv_b = (const float*)d_in[8];
    float* out = (float*)d_out;

    // Workspace layout (floats): kv[B*256*2] interleaved | kmin[B] | kmax[B]
    float* ws_kv   = (float*)d_ws;
    float* ws_kmin = ws_kv + BATCH * 2 * NKV;
    float* ws_kmax = ws_kmin + BATCH;

    gsa_prep_kernel<<<dim3(BATCH), dim3(256), 0, stream>>>(
        x, wk, bk, wv, bv, conv_w, conv_b, ws_kv, ws_kmin, ws_kmax);

    gsa_attn_kernel<<<dim3(HW / 128, BATCH), dim3(256), 0, stream>>>(
        x, wq, bq, ws_kv, ws_kmin, ws_kmax, out);
}